// HeteroRGCNLayer_76055280878116
// MI455X (gfx1250) — compile-verified
//
#include <hip/hip_runtime.h>

// ---------------------------------------------------------------------------
// HeteroRGCN layer for MI455X (gfx1250, wave32)
//
//   Wh_r = x @ W_r + b_r           (fp32 WMMA: v_wmma_f32_16x16x4_f32)
//   out  = (1/R) * sum_r segmean_r (fused as scale = 1/(R*max(cnt,1)),
//                                   applied at scatter time with f32 atomics)
//
// Pipeline (per relation, sequential on stream so Wh_r (51.2MB) stays hot in
// the 192MB L2 between its GEMM and its scatter):
//   zero(out,cnt) -> histogram(dst) -> cnt->scale -> [gemm_r -> scatter_r] x4
//
// GEMM register tiling: one wave computes a 64x64 output tile as 4x4 WMMA
// tiles -> per K-step: 4 x b64 A-loads + 8 x b32 B-loads feed 16 WMMAs
// (vs 33 vmem per 16 WMMAs for a 1x16 tile shape).
// ---------------------------------------------------------------------------

typedef __attribute__((ext_vector_type(2))) float v2f;
typedef __attribute__((ext_vector_type(8))) float v8f;

static constexpr int NND   = 50000;   // nodes
static constexpr int DIN   = 256;     // input dim
static constexpr int DOUT  = 256;     // output dim
static constexpr int RREL  = 4;       // relations
static constexpr int NEDGE = 250000;  // edges per relation

// ---- zero output accumulator and per-(r,node) counters ---------------------
__global__ void __launch_bounds__(256) rgcn_zero(float* __restrict__ out, int nOut,
                                                 int* __restrict__ cnt, int nCnt) {
    int i = blockIdx.x * 256 + threadIdx.x;
    if (i < nOut) out[i] = 0.0f;
    if (i < nCnt) cnt[i] = 0;
}

// ---- in-degree histogram over all relations: cnt[r*N + dst] ----------------
__global__ void __launch_bounds__(256) rgcn_count(const int* __restrict__ dst,
                                                  int* __restrict__ cnt,
                                                  int total, int N, int E) {
    int i = blockIdx.x * 256 + threadIdx.x;
    if (i >= total) return;
    int r = i / E;
    atomicAdd(&cnt[r * N + dst[i]], 1);
}

// ---- cnt -> scale = 1/(R * max(cnt,1)), reinterpreted in place -------------
__global__ void __launch_bounds__(256) rgcn_scale(float* __restrict__ buf,
                                                  int total, float invR) {
    int i = blockIdx.x * 256 + threadIdx.x;
    if (i >= total) return;
    int c = ((const int*)buf)[i];
    float cf = (float)(c > 1 ? c : 1);
    buf[i] = invR / cf;
}

// ---- fp32 WMMA GEMM: Wh = x @ W + bias  (one wave per 64x64 output tile) ---
// A layout (16x4 f32): lanes 0-15 -> M=lane, v0:K=0, v1:K=1
//                      lanes 16-31 -> M=lane-16, v0:K=2, v1:K=3
// B layout (4x16 f32): mirrored, indexed by N.
// C/D (16x16 f32): vgpr i, lanes 0-15 -> (M=i, N=lane); lanes 16-31 -> (M=i+8).
__global__ void __launch_bounds__(32) rgcn_gemm(const float* __restrict__ x,
                                                const float* __restrict__ W,
                                                const float* __restrict__ bias,
                                                float* __restrict__ Wh) {
    const int lane = threadIdx.x & 31;
    const int l15  = lane & 15;
    const int half = lane >> 4;              // 0: K 0/1, 1: K 2/3
    const int m0   = blockIdx.x * 64;        // 4 M-tiles of 16 rows
    const int n0   = blockIdx.y * 64;        // 4 N-tiles of 16 cols

    v8f acc[4][4];
#pragma unroll
    for (int mt = 0; mt < 4; ++mt)
#pragma unroll
        for (int nt = 0; nt < 4; ++nt)
            acc[mt][nt] = (v8f){0.f,0.f,0.f,0.f,0.f,0.f,0.f,0.f};

    // Per-M-tile A row pointers; clamp OOB rows of the tail block so every
    // lane issues a valid load (WMMA needs EXEC all-ones -> no predication).
    const float* __restrict__ xrow[4];
#pragma unroll
    for (int mt = 0; mt < 4; ++mt) {
        int row = m0 + mt * 16 + l15;
        if (row > NND - 1) row = NND - 1;
        xrow[mt] = x + (size_t)row * DIN;
    }

    for (int k0 = 0; k0 < DIN; k0 += 4) {
        const int kb = k0 + half * 2;        // this lane-half's first K

        v2f a[4];
#pragma unroll
        for (int mt = 0; mt < 4; ++mt)
            a[mt] = *(const v2f*)(xrow[mt] + kb);   // b64 load: K=kb, kb+1

        const float* __restrict__ Wk = W + (size_t)kb * DOUT + n0;
        v2f bf[4];
#pragma unroll
        for (int nt = 0; nt < 4; ++nt) {
            bf[nt].x = Wk[nt * 16 + l15];           // K = kb
            bf[nt].y = Wk[DOUT + nt * 16 + l15];    // K = kb+1
        }

#pragma unroll
        for (int mt = 0; mt < 4; ++mt)
#pragma unroll
            for (int nt = 0; nt < 4; ++nt)
                acc[mt][nt] = __builtin_amdgcn_wmma_f32_16x16x4_f32(
                    false, a[mt], false, bf[nt], (short)0, acc[mt][nt],
                    false, false);
    }

#pragma unroll
    for (int mt = 0; mt < 4; ++mt) {
        const int mbase = m0 + mt * 16;
        if (mbase >= NND) break;             // tail tiles are all-or-nothing
#pragma unroll
        for (int nt = 0; nt < 4; ++nt) {
            const int   col = n0 + nt * 16 + l15;
            const float bv  = bias[col];
#pragma unroll
            for (int i = 0; i < 8; ++i) {
                const int row = mbase + half * 8 + i;
                Wh[(size_t)row * DOUT + col] = acc[mt][nt][i] + bv;
            }
        }
    }
}

// ---- edge scatter: out[dst] += Wh[src] * scale[dst] ------------------------
// 64 threads per edge, float4 per thread (256 floats per edge message).
__global__ void __launch_bounds__(256) rgcn_scatter(const float* __restrict__ Wh,
                                                    const int* __restrict__ src,
                                                    const int* __restrict__ dst,
                                                    const float* __restrict__ scale,
                                                    float* __restrict__ out, int E) {
    long long gid = (long long)blockIdx.x * 256 + threadIdx.x;
    int e = (int)(gid >> 6);
    int q = (int)(gid & 63);
    if (e >= E) return;

    const int   s  = src[e];
    const int   d  = dst[e];
    const float sc = scale[d];

    const float4 v = *((const float4*)(Wh + (size_t)s * DOUT) + q);
    float* o = out + (size_t)d * DOUT + q * 4;
    atomicAdd(o + 0, v.x * sc);
    atomicAdd(o + 1, v.y * sc);
    atomicAdd(o + 2, v.z * sc);
    atomicAdd(o + 3, v.w * sc);
}

// ---------------------------------------------------------------------------
extern "C" void kernel_launch(void* const* d_in, const int* in_sizes, int n_in,
                              void* d_out, int out_size, void* d_ws, size_t ws_size,
                              hipStream_t stream) {
    (void)in_sizes; (void)n_in; (void)out_size; (void)ws_size;

    const float* x    = (const float*)d_in[0];   // [N, DIN]
    const float* W    = (const float*)d_in[1];   // [R, DIN, DOUT]
    const float* bias = (const float*)d_in[2];   // [R, DOUT]
    const int*   src  = (const int*)d_in[3];     // [R, E]
    const int*   dst  = (const int*)d_in[4];     // [R, E]
    float*       out  = (float*)d_out;           // [N, DOUT]

    // workspace: [R*N floats scale/cnt][N*DOUT floats Wh]  (~52 MB)
    float* scaleBuf = (float*)d_ws;
    int*   cnt      = (int*)d_ws;
    float* Wh       = (float*)((char*)d_ws + (size_t)RREL * NND * sizeof(float));

    const int nOut = NND * DOUT;       // 12.8M
    const int nCnt = RREL * NND;       // 200k

    rgcn_zero<<<(nOut + 255) / 256, 256, 0, stream>>>(out, nOut, cnt, nCnt);
    rgcn_count<<<(RREL * NEDGE + 255) / 256, 256, 0, stream>>>(
        dst, cnt, RREL * NEDGE, NND, NEDGE);
    rgcn_scale<<<(nCnt + 255) / 256, 256, 0, stream>>>(
        scaleBuf, nCnt, 1.0f / (float)RREL);

    const int mtiles  = (NND + 15) / 16;         // 3125
    const int mblocks = (mtiles + 3) / 4;        // 782 (last block is partial)
    dim3 ggrid(mblocks, DOUT / 64);              // 782 x 4 waves

    for (int r = 0; r < RREL; ++r) {
        rgcn_gemm<<<ggrid, 32, 0, stream>>>(
            x, W + (size_t)r * DIN * DOUT, bias + (size_t)r * DOUT, Wh);
        rgcn_scatter<<<(NEDGE * 64) / 256, 256, 0, stream>>>(
            Wh, src + (size_t)r * NEDGE, dst + (size_t)r * NEDGE,
            scaleBuf + (size_t)r * NND, out, NEDGE);
    }
}